// QFF_55791625175294
// MI455X (gfx1250) — compile-verified
//
#include <hip/hip_runtime.h>

// QFF frequency-grid trilinear interpolation, tuned for MI455X (gfx1250).
//  - wave32: lane == group (G = 32 = NUM_FREQS*2)
//  - one wave per point, grid-stride loop
//  - table (64 MB) resident in 192 MB L2; outputs use non-temporal stores
//    so the 536 MB output stream does not evict the table
//  - global_prefetch_b8 for the next point row

constexpr int kQ   = 64;
constexpr int kQ2  = kQ * kQ;
constexpr int kQ3  = kQ * kQ * kQ;
constexpr int kG   = 32;                 // NUM_FREQS * 2 == wave32 width
constexpr int kC   = 2;                  // NUM_FEATS
constexpr int kOutStride = 3 + kG * kC;  // 67 floats per output row

#define INV_TWO_PI 0.15915494309189535f

__global__ __launch_bounds__(256) void qff_kernel(
    const float* __restrict__ points,   // (N, 3)
    const float* __restrict__ freqs,    // (16,)
    const float* __restrict__ cv,       // (32, 2, 64, 64, 64)
    float* __restrict__ out,            // (N, 67)
    int n_pts)
{
    const int lane          = threadIdx.x & 31;
    const int waveInBlock   = threadIdx.x >> 5;
    const int wavesPerBlock = blockDim.x >> 5;
    const int wave          = blockIdx.x * wavesPerBlock + waveInBlock;
    const int totalWaves    = gridDim.x * wavesPerBlock;

    // lane -> (freq index, sin/cos). group g == lane.
    const int   f      = lane >> 1;
    const float qoff   = (lane & 1) ? 0.25f : 0.0f;      // cos(x) = sin(x + 1/4 turn)
    const float fscale = freqs[f] * INV_TWO_PI;          // phase -> revolutions
    const float* __restrict__ cvg = cv + (size_t)lane * (size_t)(kC * kQ3);

    for (int n = wave; n < n_pts; n += totalWaves) {
        // Prefetch next point row (global_prefetch_b8); speculative, cheap.
        const int np = n + totalWaves;
        if (np < n_pts) __builtin_prefetch(&points[3 * (size_t)np], 0, 3);

        const size_t pb = 3 * (size_t)n;
        const float px = points[pb + 0];
        const float py = points[pb + 1];
        const float pz = points[pb + 2];

        // coords = sin/cos(point * freq) via hw v_sin_f32 (input in revolutions)
        const float c0 = __builtin_amdgcn_sinf(__builtin_fmaf(px, fscale, qoff)); // -> z
        const float c1 = __builtin_amdgcn_sinf(__builtin_fmaf(py, fscale, qoff)); // -> y
        const float c2 = __builtin_amdgcn_sinf(__builtin_fmaf(pz, fscale, qoff)); // -> x

        // map [-1,1] -> [0, Q-1]
        const float sc = 0.5f * (float)(kQ - 1);
        const float xz = (c0 + 1.0f) * sc;
        const float xy = (c1 + 1.0f) * sc;
        const float xx = (c2 + 1.0f) * sc;

        const float fz = floorf(xz), fy = floorf(xy), fx = floorf(xx);
        // fractional weights BEFORE index clamping (matches reference exactly)
        const float wz = xz - fz, wy = xy - fy, wx = xx - fx;

        const int iz0 = min(max((int)fz, 0), kQ - 2);
        const int iy0 = min(max((int)fy, 0), kQ - 2);
        const int ix0 = min(max((int)fx, 0), kQ - 2);
        const int iz1 = iz0 + 1, iy1 = iy0 + 1, ix1 = ix0 + 1;

        const float wz0 = 1.0f - wz, wy0 = 1.0f - wy, wx0 = 1.0f - wx;

        // flat offsets for the 8 corners (channel 0 plane)
        const int b00 = iz0 * kQ2 + iy0 * kQ;
        const int b01 = iz0 * kQ2 + iy1 * kQ;
        const int b10 = iz1 * kQ2 + iy0 * kQ;
        const int b11 = iz1 * kQ2 + iy1 * kQ;

        const int o000 = b00 + ix0, o001 = b00 + ix1;
        const int o010 = b01 + ix0, o011 = b01 + ix1;
        const int o100 = b10 + ix0, o101 = b10 + ix1;
        const int o110 = b11 + ix0, o111 = b11 + ix1;

        // issue all 16 gathers (L2-resident table); compiler batches & overlaps
        const float a000 = cvg[o000], a001 = cvg[o001];
        const float a010 = cvg[o010], a011 = cvg[o011];
        const float a100 = cvg[o100], a101 = cvg[o101];
        const float a110 = cvg[o110], a111 = cvg[o111];
        const float e000 = cvg[kQ3 + o000], e001 = cvg[kQ3 + o001];
        const float e010 = cvg[kQ3 + o010], e011 = cvg[kQ3 + o011];
        const float e100 = cvg[kQ3 + o100], e101 = cvg[kQ3 + o101];
        const float e110 = cvg[kQ3 + o110], e111 = cvg[kQ3 + o111];

        // trilinear weights
        const float w00 = wz0 * wy0, w01 = wz0 * wy, w10 = wz * wy0, w11 = wz * wy;
        const float w000 = w00 * wx0, w001 = w00 * wx;
        const float w010 = w01 * wx0, w011 = w01 * wx;
        const float w100 = w10 * wx0, w101 = w10 * wx;
        const float w110 = w11 * wx0, w111 = w11 * wx;

        float acc0 = a000 * w000;
        acc0 = __builtin_fmaf(a001, w001, acc0);
        acc0 = __builtin_fmaf(a010, w010, acc0);
        acc0 = __builtin_fmaf(a011, w011, acc0);
        acc0 = __builtin_fmaf(a100, w100, acc0);
        acc0 = __builtin_fmaf(a101, w101, acc0);
        acc0 = __builtin_fmaf(a110, w110, acc0);
        acc0 = __builtin_fmaf(a111, w111, acc0);

        float acc1 = e000 * w000;
        acc1 = __builtin_fmaf(e001, w001, acc1);
        acc1 = __builtin_fmaf(e010, w010, acc1);
        acc1 = __builtin_fmaf(e011, w011, acc1);
        acc1 = __builtin_fmaf(e100, w100, acc1);
        acc1 = __builtin_fmaf(e101, w101, acc1);
        acc1 = __builtin_fmaf(e110, w110, acc1);
        acc1 = __builtin_fmaf(e111, w111, acc1);

        // out[n, 3 + 2*lane + {0,1}] — non-temporal: don't evict the L2-resident table
        const size_t ob = (size_t)n * kOutStride;
        float* __restrict__ orow = out + ob + 3 + 2 * lane;
        __builtin_nontemporal_store(acc0, orow + 0);
        __builtin_nontemporal_store(acc1, orow + 1);

        // passthrough of the 3 point coords (lanes 0..2)
        if (lane < 3) {
            const float v = (lane == 0) ? px : (lane == 1) ? py : pz;
            __builtin_nontemporal_store(v, out + ob + lane);
        }
    }
}

extern "C" void kernel_launch(void* const* d_in, const int* in_sizes, int n_in,
                              void* d_out, int out_size, void* d_ws, size_t ws_size,
                              hipStream_t stream) {
    const float* points = (const float*)d_in[0];
    const float* freqs  = (const float*)d_in[1];
    const float* cv     = (const float*)d_in[2];
    float* out = (float*)d_out;

    const int n_pts = in_sizes[0] / 3;

    // 256 threads = 8 wave32 per block; one point per wave per iteration.
    const dim3 block(256);
    int blocks = 4096;                       // 32768 waves in flight, grid-stride
    const int maxBlocks = (n_pts + 7) / 8;   // never more waves than points
    if (blocks > maxBlocks) blocks = maxBlocks;
    if (blocks < 1) blocks = 1;

    qff_kernel<<<blocks, block, 0, stream>>>(points, freqs, cv, out, n_pts);
}